// Encode_decode_atten_70557722739407
// MI455X (gfx1250) — compile-verified
//
#include <hip/hip_runtime.h>
#include <hip/hip_bf16.h>

#define VOCAB 100000
#define E     256
#define H     128
#define G3    384      // 3*H
#define SEQ   4096
#define DEC   10

typedef float v2f __attribute__((ext_vector_type(2)));
typedef float v8f __attribute__((ext_vector_type(8)));

__device__ __forceinline__ float sigm(float x) { return 1.0f / (1.0f + expf(-x)); }

// ---------------------------------------------------------------------------
// K1: embedding gather  X[t, :] = embed[ids[t], :]
// ---------------------------------------------------------------------------
__global__ void gather_embed(const int* __restrict__ ids,
                             const float* __restrict__ embed,
                             float* __restrict__ X) {
    int t = blockIdx.x;
    int e = threadIdx.x;
    X[(size_t)t * E + e] = embed[(size_t)ids[t] * E + e];
}

// ---------------------------------------------------------------------------
// K2/K4: C[M,N] = A[M,K] * B^T + bias   (B stored row-major as (N, ldb), so
// matrix element B[k][n] = B[n*ldb + k]).  One wave32 per 16x16 output tile,
// K-loop of V_WMMA_F32_16X16X4_F32 (full fp32 precision).
// ---------------------------------------------------------------------------
__global__ __launch_bounds__(256) void wmma_gemm_xwT(
    const float* __restrict__ A, const float* __restrict__ B,
    const float* __restrict__ bias, float* __restrict__ C,
    int M, int N, int K, int lda, int ldb, int ldc) {
    int gthread = blockIdx.x * blockDim.x + threadIdx.x;
    int wave    = gthread >> 5;
    int lane    = threadIdx.x & 31;
    int tilesN  = N >> 4;
    int tm      = (wave / tilesN) << 4;
    int tn      = (wave % tilesN) << 4;
    if (tm >= M) return;                       // wave-uniform: EXEC stays all-1s

    int l16 = lane & 15;
    int hi  = (lane >> 4) & 1;                 // 0: lanes 0-15, 1: lanes 16-31

    // A 16x4 f32 fragment: lane l16 = row; VGPR0/1 hold K = 2*hi, 2*hi+1
    const float* Arow = A + (size_t)(tm + l16) * lda + 2 * hi;
    // B 4x16 f32 fragment (mirrored): lane l16 = col n; K = 2*hi, 2*hi+1
    const float* Brow = B + (size_t)(tn + l16) * ldb + 2 * hi;

    v8f c = {};
    for (int k0 = 0; k0 < K; k0 += 4) {
        v2f a, b;
        a.x = Arow[k0];
        a.y = Arow[k0 + 1];
        b.x = Brow[k0];
        b.y = Brow[k0 + 1];
        c = __builtin_amdgcn_wmma_f32_16x16x4_f32(false, a, false, b,
                                                  (short)0, c, false, false);
    }

    // C/D 16x16 f32: VGPR v -> row tm + v + 8*hi, col tn + l16
    int   n  = tn + l16;
    float bv = bias[n];
#pragma unroll
    for (int v = 0; v < 8; ++v) {
        int m = tm + v + 8 * hi;
        C[(size_t)m * ldc + n] = c[v] + bv;
    }
}

// ---------------------------------------------------------------------------
// K3: sequential encoder GRU scan. Single workgroup, 384 threads (12 waves).
// W_hh kept transposed in LDS (128x384 f32 = 192KB; CDNA5 WGP has 320KB LDS)
// so lane accesses are consecutive -> bank-conflict-free.
// dyn smem: WhhT[H*G3] | gh[G3] | hbuf[H]
// ---------------------------------------------------------------------------
__global__ __launch_bounds__(G3) void enc_scan(
    const float* __restrict__ Gi, const float* __restrict__ Whh,
    const float* __restrict__ b_hh, float* __restrict__ oute,
    float* __restrict__ h_final) {
    extern __shared__ float smem[];
    float* WhhT = smem;           // H*G3 floats
    float* gh   = WhhT + H * G3;  // G3 floats
    float* hbuf = gh + G3;        // H floats

    int j = threadIdx.x;          // 0..383
    // load W_hh transposed: WhhT[k*G3 + j] = Whh[j*H + k]
    for (int k = 0; k < H; ++k) WhhT[k * G3 + j] = Whh[j * H + k];
    if (j < H) hbuf[j] = 0.0f;
    __syncthreads();

    float bh = b_hh[j];
    for (int t = 0; t < SEQ; ++t) {
        // gh[j] = b_hh[j] + dot(W_hh[j,:], h)
        float s = bh;
#pragma unroll 8
        for (int k = 0; k < H; ++k) s += WhhT[k * G3 + j] * hbuf[k];
        gh[j] = s;
        __syncthreads();

        if (j < H) {
            const float* gi = Gi + (size_t)t * G3;
            float r  = sigm(gi[j]         + gh[j]);
            float z  = sigm(gi[H + j]     + gh[H + j]);
            float n  = tanhf(gi[2 * H + j] + r * gh[2 * H + j]);
            float hn = (1.0f - z) * n + z * hbuf[j];
            oute[(size_t)t * H + j] = hn;
            hbuf[j] = hn;                     // only thread j touches hbuf[j]
        }
        __syncthreads();
    }
    if (j < H) h_final[j] = hbuf[j];
}

// ---------------------------------------------------------------------------
// K4-pre: fold the h_final half of the FC into a per-column constant:
// c2[i] = b_fc[i] + dot(h_final, W_fc[i, H:2H])
// ---------------------------------------------------------------------------
__global__ void fc_const(const float* __restrict__ h_final,
                         const float* __restrict__ W_fc,
                         const float* __restrict__ b_fc,
                         float* __restrict__ c2) {
    int i = threadIdx.x;
    if (i >= H) return;
    float s = b_fc[i];
    for (int k = 0; k < H; ++k) s += h_final[k] * W_fc[(size_t)i * (2 * H) + H + k];
    c2[i] = s;
}

// ---------------------------------------------------------------------------
// K5: per-column softmax over seq dim + weighted reduction with oute.
// colctx[i] = sum_t softmax(scores[:,i])[t] * oute[t][i]
// ---------------------------------------------------------------------------
__global__ __launch_bounds__(256) void softmax_col(
    const float* __restrict__ scores, const float* __restrict__ oute,
    float* __restrict__ colctx) {
    __shared__ float red[256];
    __shared__ float red2[256];
    int i   = blockIdx.x;
    int tid = threadIdx.x;

    float m = -1e30f;
    for (int t = tid; t < SEQ; t += 256) m = fmaxf(m, scores[(size_t)t * H + i]);
    red[tid] = m;
    __syncthreads();
    for (int s = 128; s > 0; s >>= 1) {
        if (tid < s) red[tid] = fmaxf(red[tid], red[tid + s]);
        __syncthreads();
    }
    m = red[0];
    __syncthreads();

    float se = 0.0f, ce = 0.0f;
    for (int t = tid; t < SEQ; t += 256) {
        float e = expf(scores[(size_t)t * H + i] - m);
        se += e;
        ce += e * oute[(size_t)t * H + i];
    }
    red[tid]  = se;
    red2[tid] = ce;
    __syncthreads();
    for (int s = 128; s > 0; s >>= 1) {
        if (tid < s) { red[tid] += red[tid + s]; red2[tid] += red2[tid + s]; }
        __syncthreads();
    }
    if (tid == 0) colctx[i] = red2[0] / red[0];
}

// ---------------------------------------------------------------------------
// K6: context = sum(colctx); 10-step scalar decoder GRUCell(1,1); sigmoid.
// ---------------------------------------------------------------------------
__global__ void finalize(const float* __restrict__ colctx,
                         const float* __restrict__ dW_ih,
                         const float* __restrict__ dW_hh,
                         const float* __restrict__ db_ih,
                         const float* __restrict__ db_hh,
                         float* __restrict__ out) {
    if (threadIdx.x != 0 || blockIdx.x != 0) return;
    float ctx = 0.0f;
    for (int i = 0; i < H; ++i) ctx += colctx[i];

    float h = 0.0f;
    float x = ctx;                       // step 1: cell(context, 0); then cell(h, h)
    for (int step = 0; step < DEC; ++step) {
        float gr = x * dW_ih[0] + db_ih[0] + h * dW_hh[0] + db_hh[0];
        float gz = x * dW_ih[1] + db_ih[1] + h * dW_hh[1] + db_hh[1];
        float gn = x * dW_ih[2] + db_ih[2];
        float hh = h * dW_hh[2] + db_hh[2];
        float r  = sigm(gr);
        float z  = sigm(gz);
        float n  = tanhf(gn + r * hh);
        h = (1.0f - z) * n + z * h;
        x = h;
    }
    out[0] = sigm(h);
}

// ---------------------------------------------------------------------------
extern "C" void kernel_launch(void* const* d_in, const int* in_sizes, int n_in,
                              void* d_out, int out_size, void* d_ws, size_t ws_size,
                              hipStream_t stream) {
    const int*   ids   = (const int*)d_in[0];
    const float* embed = (const float*)d_in[1];
    const float* W_ih  = (const float*)d_in[2];
    const float* W_hh  = (const float*)d_in[3];
    const float* b_ih  = (const float*)d_in[4];
    const float* b_hh  = (const float*)d_in[5];
    const float* W_fc  = (const float*)d_in[6];
    const float* b_fc  = (const float*)d_in[7];
    const float* dW_ih = (const float*)d_in[8];
    const float* dW_hh = (const float*)d_in[9];
    const float* db_ih = (const float*)d_in[10];
    const float* db_hh = (const float*)d_in[11];
    (void)in_sizes; (void)n_in; (void)out_size; (void)ws_size;

    float* ws      = (float*)d_ws;
    float* X       = ws;                    // SEQ*E
    float* Gi      = X + (size_t)SEQ * E;   // SEQ*G3
    float* oute    = Gi + (size_t)SEQ * G3; // SEQ*H
    float* scores  = oute + (size_t)SEQ * H;// SEQ*H
    float* h_final = scores + (size_t)SEQ * H; // H
    float* c2      = h_final + H;           // H
    float* colctx  = c2 + H;                // H

    // K1: gather embeddings
    gather_embed<<<SEQ, E, 0, stream>>>(ids, embed, X);

    // K2: Gi = X(4096x256) @ W_ih^T(256x384) + b_ih   [WMMA f32]
    {
        int waves  = (SEQ / 16) * (G3 / 16);      // 6144
        int blocks = waves * 32 / 256;            // 768
        wmma_gemm_xwT<<<blocks, 256, 0, stream>>>(X, W_ih, b_ih, Gi,
                                                  SEQ, G3, E, E, E, G3);
    }

    // K3: sequential GRU scan (1 WGP, 192KB W_hh^T in LDS)
    {
        size_t smem = (size_t)(H * G3 + G3 + H) * sizeof(float);  // 198656 B
        enc_scan<<<1, G3, smem, stream>>>(Gi, W_hh, b_hh, oute, h_final);
    }

    // K4-pre: per-column constant from h_final half of FC
    fc_const<<<1, H, 0, stream>>>(h_final, W_fc, b_fc, c2);

    // K4: scores = oute(4096x128) @ W_fc[:, :128]^T + c2   [WMMA f32]
    {
        int waves  = (SEQ / 16) * (H / 16);       // 2048
        int blocks = waves * 32 / 256;            // 256
        wmma_gemm_xwT<<<blocks, 256, 0, stream>>>(oute, W_fc, c2, scores,
                                                  SEQ, H, H, H, 2 * H, H);
    }

    // K5: column softmax + weighted sum with oute
    softmax_col<<<H, 256, 0, stream>>>(scores, oute, colctx);

    // K6: context reduction + scalar decoder + sigmoid
    finalize<<<1, 1, 0, stream>>>(colctx, dW_ih, dW_hh, db_ih, db_hh, (float*)d_out);
}